// RecurrentLIF_39178691674942
// MI455X (gfx1250) — compile-verified
//
#include <hip/hip_runtime.h>
#include <stdint.h>

// RecurrentLIF on gfx1250 (MI455X): f32 WMMA GEMMs + fused LIF epilogue.
//   Phase 1: H = X @ W_lin^T + b           (one big 4096^3 fp32 GEMM)
//   Phase 2: per-timestep  mem = mem*TAU + H_t + spike @ W_rec ; spike = mem>1 ; hard reset
// Matrix math via V_WMMA_F32_16X16X4_F32 (wave32). Tiles staged to LDS with the
// CDNA5 async global->LDS path (GLOBAL_LOAD_ASYNC_TO_LDS_B128 / ASYNCcnt),
// double-buffered to overlap HBM with WMMA.

typedef __attribute__((ext_vector_type(2))) float v2f;
typedef __attribute__((ext_vector_type(8))) float v8f;
typedef __attribute__((ext_vector_type(4))) int   v4i;

#define TAU_C    0.5f
#define THRESH_C 1.0f

constexpr int Bn   = 128;   // batch
constexpr int Tn   = 32;    // time steps
constexpr int Kdim = 4096;  // reduction dim (IN == OUT)
constexpr int Ndim = 4096;  // output features

constexpr int BM = 128;     // macro-tile M
constexpr int BN = 64;      // macro-tile N
constexpr int KT = 32;      // K chunk staged in LDS
constexpr int LDA = 36;     // padded LDS stride, K-contiguous tiles (mult of 4 -> b128-aligned)

#ifndef __has_builtin
#define __has_builtin(x) 0
#endif

#if __has_builtin(__builtin_amdgcn_global_load_async_to_lds_b128)
#define USE_ASYNC 1
#else
#define USE_ASYNC 0
#endif

// 16-byte global -> LDS copy (async DMA when available, else via VGPRs)
__device__ __forceinline__ void cp16(const float* __restrict__ g, float* l)
{
#if USE_ASYNC
    // prototype (from compiler diagnostic): (v4i* global_src, v4i* lds_dst, imm offset, imm cpol)
    __builtin_amdgcn_global_load_async_to_lds_b128((v4i*)g, (v4i*)l, 0, 0);
#else
    *(float4*)l = *(const float4*)g;
#endif
}

template <int N>
__device__ __forceinline__ void wait_stage()
{
#if USE_ASYNC
#if __has_builtin(__builtin_amdgcn_s_wait_asynccnt)
    __builtin_amdgcn_s_wait_asynccnt(N);
#else
    asm volatile("s_wait_asynccnt %0" ::"i"(N) : "memory");
#endif
#endif
}

// MODE 0: C = A @ B^T + bias   (A = X [M,K] rowmajor, B = W_lin [N,K] rowmajor)
//         LDS B tile stored n-major/K-contiguous -> direct copy AND b64 fragment loads.
// MODE 1: fused LIF step       (A = spike_prev [128,K], B = W_rec [K,N] rowmajor)
//         LDS B tile stored k-major -> direct copy; fragments via 2x b32.
template <int MODE>
__global__ __launch_bounds__(256)
void wmma_gemm_lif(const float* __restrict__ Ag,
                   const float* __restrict__ Bg,
                   const float* __restrict__ bias,    // MODE 0 only
                   const float* __restrict__ Hbuf,    // MODE 1 only
                   float* __restrict__ mem,           // MODE 1 only
                   float* __restrict__ spk_next,      // MODE 1 only
                   float* __restrict__ Cg,            // MODE 0: H ; MODE 1: spikes out [B,T,N]
                   int tstep)
{
    constexpr int BROWS = (MODE == 0) ? BN : KT;
    constexpr int BLD   = (MODE == 0) ? LDA : 68;

    __shared__ float lds_a[2][BM][LDA];
    __shared__ float lds_b[2][BROWS][BLD];

    const int tid  = threadIdx.x;
    const int lane = tid & 31;
    const int wave = tid >> 5;
    const int half = lane >> 4;   // 0: lanes 0-15, 1: lanes 16-31 (K pair select)
    const int l16  = lane & 15;

    const int wm = (wave & 3) * 32;   // 4 waves along M
    const int wn = (wave >> 2) * 32;  // 2 waves along N

    const int n0 = blockIdx.x * BN;
    const int m0 = blockIdx.y * BM;

    v8f c[2][2] = {};

    // stage one KT-wide tile pair into LDS buffer `buf` (6 x b128 per thread)
    auto stage = [&](int k0, int buf) {
        const int arow = tid >> 3;          // 0..31
        const int acol = (tid & 7) << 2;    // 0,4,...,28
#pragma unroll
        for (int i = 0; i < 4; ++i)
            cp16(Ag + (size_t)(m0 + arow + i * 32) * Kdim + k0 + acol,
                 &lds_a[buf][arow + i * 32][acol]);
        if (MODE == 0) {
            const int bn = tid >> 3;
            const int bq = (tid & 7) << 2;
#pragma unroll
            for (int i = 0; i < 2; ++i)
                cp16(Bg + (size_t)(n0 + bn + i * 32) * Kdim + k0 + bq,
                     &lds_b[buf][bn + i * 32][bq]);
        } else {
            const int bk = tid >> 4;            // 0..15
            const int bc = (tid & 15) << 2;     // 0,4,...,60
#pragma unroll
            for (int i = 0; i < 2; ++i)
                cp16(Bg + (size_t)(k0 + bk + i * 16) * Ndim + n0 + bc,
                     &lds_b[buf][bk + i * 16][bc]);
        }
    };

    constexpr int NT = Kdim / KT;   // 128 K-tiles
    stage(0, 0);

    for (int kt = 0; kt < NT; ++kt) {
        const int cur = kt & 1;
        if (kt + 1 < NT) {
            stage((kt + 1) * KT, cur ^ 1);  // prefetch next tile into other buffer
            wait_stage<6>();                // current tile's 6 copies done (in-order)
        } else {
            wait_stage<0>();
        }
        __syncthreads();

#pragma unroll
        for (int ks = 0; ks < KT / 4; ++ks) {
            const int kk = ks * 4 + half * 2;   // lanes 0-15 -> K0,K1 ; 16-31 -> K2,K3
            v2f a0 = *(const v2f*)&lds_a[cur][wm + l16][kk];
            v2f a1 = *(const v2f*)&lds_a[cur][wm + 16 + l16][kk];
            v2f b0, b1;
            if (MODE == 0) {
                b0 = *(const v2f*)&lds_b[cur][wn + l16][kk];
                b1 = *(const v2f*)&lds_b[cur][wn + 16 + l16][kk];
            } else {
                b0.x = lds_b[cur][kk][wn + l16];        b0.y = lds_b[cur][kk + 1][wn + l16];
                b1.x = lds_b[cur][kk][wn + 16 + l16];   b1.y = lds_b[cur][kk + 1][wn + 16 + l16];
            }
            c[0][0] = __builtin_amdgcn_wmma_f32_16x16x4_f32(false, a0, false, b0, (short)0, c[0][0], false, false);
            c[0][1] = __builtin_amdgcn_wmma_f32_16x16x4_f32(false, a0, false, b1, (short)0, c[0][1], false, false);
            c[1][0] = __builtin_amdgcn_wmma_f32_16x16x4_f32(false, a1, false, b0, (short)0, c[1][0], false, false);
            c[1][1] = __builtin_amdgcn_wmma_f32_16x16x4_f32(false, a1, false, b1, (short)0, c[1][1], false, false);
        }
        __syncthreads();
    }

    // epilogue: C/D layout = VGPR r, lanes 0-15 -> M=r, lanes 16-31 -> M=r+8, N=lane&15
#pragma unroll
    for (int i = 0; i < 2; ++i) {
#pragma unroll
        for (int j = 0; j < 2; ++j) {
            const int col = n0 + wn + j * 16 + l16;
            if (MODE == 0) {
                const float bv = bias[col];
#pragma unroll
                for (int r = 0; r < 8; ++r) {
                    const int m = m0 + wm + i * 16 + half * 8 + r;
                    Cg[(size_t)m * Ndim + col] = c[i][j][r] + bv;
                }
            } else {
#pragma unroll
                for (int r = 0; r < 8; ++r) {
                    const int b  = m0 + wm + i * 16 + half * 8 + r;          // batch index
                    const size_t bo = (size_t)b * Ndim + col;                // [B,N] state
                    const size_t ho = (size_t)(b * Tn + tstep) * Ndim + col; // [B,T,N]
                    const float mm = mem[bo] * TAU_C + Hbuf[ho] + c[i][j][r];
                    const float sp = (mm > THRESH_C) ? 1.0f : 0.0f;
                    mem[bo]      = (1.0f - sp) * mm;   // hard reset on spike
                    spk_next[bo] = sp;
                    Cg[ho]       = sp;
                }
            }
        }
    }
}

// t = 0: spike_prev = 0, mem_prev = 0  ->  mem = H_0, pure elementwise.
__global__ __launch_bounds__(256)
void lif_first(const float* __restrict__ Hbuf, float* __restrict__ mem,
               float* __restrict__ spk, float* __restrict__ out)
{
    const int idx = blockIdx.x * blockDim.x + threadIdx.x;   // over Bn*Ndim
    const int b = idx >> 12;           // /4096
    const int n = idx & (Ndim - 1);
    const size_t ho = (size_t)(b * Tn + 0) * Ndim + n;
    const float mm = Hbuf[ho];
    const float sp = (mm > THRESH_C) ? 1.0f : 0.0f;
    mem[idx] = (1.0f - sp) * mm;
    spk[idx] = sp;
    out[ho]  = sp;
}

extern "C" void kernel_launch(void* const* d_in, const int* in_sizes, int n_in,
                              void* d_out, int out_size, void* d_ws, size_t ws_size,
                              hipStream_t stream)
{
    const float* x     = (const float*)d_in[0];  // [128,32,4096]
    const float* W_lin = (const float*)d_in[1];  // [4096,4096] (OUT,IN)
    const float* bias  = (const float*)d_in[2];  // [4096]
    const float* W_rec = (const float*)d_in[3];  // [4096,4096] (OUT,OUT)
    float* out = (float*)d_out;                  // spikes [128,32,4096]

    // workspace layout: H (64MB) | mem (2MB) | spkA (2MB) | spkB (2MB)
    float* H    = (float*)d_ws;
    float* mem  = H    + (size_t)Bn * Tn * Ndim;
    float* spkA = mem  + (size_t)Bn * Ndim;
    float* spkB = spkA + (size_t)Bn * Ndim;

    const dim3 blk(256);

    // Phase 1: H = X @ W_lin^T + b   (4096x4096x4096)
    const dim3 g1(Ndim / BN, (Bn * Tn) / BM);   // 64 x 32 workgroups
    wmma_gemm_lif<0><<<g1, blk, 0, stream>>>(x, W_lin, bias, nullptr, nullptr, nullptr, H, 0);

    // Phase 2, t = 0 (spike/mem start at zero)
    lif_first<<<(Bn * Ndim) / 256, blk, 0, stream>>>(H, mem, spkA, out);

    // Phase 2, t = 1..31: GEMM over spikes + fused LIF update (ping-pong spike buffers)
    const dim3 g2(Ndim / BN, Bn / BM);          // 64 x 1 workgroups
    for (int t = 1; t < Tn; ++t) {
        float* sprev = (t & 1) ? spkA : spkB;
        float* snext = (t & 1) ? spkB : spkA;
        wmma_gemm_lif<1><<<g2, blk, 0, stream>>>(sprev, W_rec, nullptr, H, mem, snext, out, t);
    }
}